// MetaNCA_86474871537726
// MI455X (gfx1250) — compile-verified
//
#include <hip/hip_runtime.h>
#include <math.h>

typedef float v2f __attribute__((ext_vector_type(2)));
typedef float v8f __attribute__((ext_vector_type(8)));

#define IN_  3072
#define OUT_ 3072
#define B_   2048
#define H_   10
#define D_   6145   // IN + OUT + 1

// ---------------------------------------------------------------------------
// K1a: row_part[i][h] = sum_k weight[i][k] * W1[h][k]      (W1a = W1[:, :3072])
// one wave per 4 rows; lanes stride K, cross-lane shuffle reduce (wave32)
// ---------------------------------------------------------------------------
__global__ __launch_bounds__(256) void k_rowpart(const float* __restrict__ weight,
                                                 const float* __restrict__ W1,
                                                 float* __restrict__ rp) {
    const int wave = threadIdx.x >> 5;
    const int lane = threadIdx.x & 31;
    const int rbase = (blockIdx.x * 8 + wave) * 4;
    for (int r = 0; r < 4; ++r) {
        const int row = rbase + r;
        float acc[H_];
#pragma unroll
        for (int h = 0; h < H_; ++h) acc[h] = 0.f;
        for (int k = lane; k < IN_; k += 32) {
            const float x = weight[(size_t)row * IN_ + k];
#pragma unroll
            for (int h = 0; h < H_; ++h)
                acc[h] = fmaf(x, W1[h * D_ + k], acc[h]);
        }
#pragma unroll
        for (int h = 0; h < H_; ++h) {
            float v = acc[h];
            for (int off = 16; off > 0; off >>= 1) v += __shfl_down(v, off);
            if (lane == 0) rp[row * H_ + h] = v;
        }
    }
}

// ---------------------------------------------------------------------------
// K1b: cb[j][h] = sum_i weight[i][j]*W1b[h][i] + bias[j]*W1c[h] + b1[h]
// ---------------------------------------------------------------------------
__global__ __launch_bounds__(256) void k_colbias(const float* __restrict__ weight,
                                                 const float* __restrict__ bias,
                                                 const float* __restrict__ W1,
                                                 const float* __restrict__ b1,
                                                 float* __restrict__ cb) {
    const int j = blockIdx.x * 256 + threadIdx.x;
    float acc[H_];
#pragma unroll
    for (int h = 0; h < H_; ++h) acc[h] = 0.f;
    for (int i = 0; i < IN_; ++i) {
        const float w = weight[(size_t)i * OUT_ + j];
#pragma unroll
        for (int h = 0; h < H_; ++h)
            acc[h] = fmaf(w, W1[h * D_ + OUT_ + i], acc[h]);
    }
    const float bj = bias[j];
#pragma unroll
    for (int h = 0; h < H_; ++h)
        cb[j * H_ + h] = acc[h] + bj * W1[h * D_ + (D_ - 1)] + b1[h];
}

// ---------------------------------------------------------------------------
// K2: new_w[i][j] = weight[i][j] + W3 . relu(W2 . relu(rp[i]+cb[j]) + b2) + b3
// ---------------------------------------------------------------------------
__global__ __launch_bounds__(256) void k_update(const float* __restrict__ weight,
                                                const float* __restrict__ rp,
                                                const float* __restrict__ cb,
                                                const float* __restrict__ W2,
                                                const float* __restrict__ b2,
                                                const float* __restrict__ W3,
                                                const float* __restrict__ b3,
                                                float* __restrict__ nw) {
    __shared__ float rps[16 * H_], cbs[16 * H_], W2s[H_ * H_], b2s[H_], W3s[H_];
    const int t = threadIdx.x;
    const int i0 = blockIdx.y * 16, j0 = blockIdx.x * 16;
    if (t < 16 * H_) { rps[t] = rp[i0 * H_ + t]; cbs[t] = cb[j0 * H_ + t]; }
    if (t < H_ * H_) W2s[t] = W2[t];
    if (t < H_) { b2s[t] = b2[t]; W3s[t] = W3[t]; }
    __syncthreads();

    const int ti = t >> 4, tj = t & 15;
    float h1[H_];
#pragma unroll
    for (int h = 0; h < H_; ++h)
        h1[h] = fmaxf(rps[ti * H_ + h] + cbs[tj * H_ + h], 0.f);
    float upd = b3[0];
#pragma unroll
    for (int g = 0; g < H_; ++g) {
        float v = b2s[g];
#pragma unroll
        for (int h = 0; h < H_; ++h) v = fmaf(W2s[g * H_ + h], h1[h], v);
        upd = fmaf(W3s[g], fmaxf(v, 0.f), upd);
    }
    const size_t idx = (size_t)(i0 + ti) * OUT_ + (j0 + tj);
    nw[idx] = weight[idx] + upd;
}

// ---------------------------------------------------------------------------
// K3: g-path -> new_b[j]
// ---------------------------------------------------------------------------
__global__ __launch_bounds__(256) void k_gpath(const float* __restrict__ nw,
                                               const float* __restrict__ bias,
                                               const float* __restrict__ W1,
                                               const float* __restrict__ b1,
                                               const float* __restrict__ W2,
                                               const float* __restrict__ b2,
                                               const float* __restrict__ W3,
                                               const float* __restrict__ b3,
                                               float* __restrict__ nb) {
    const int j = blockIdx.x * 256 + threadIdx.x;
    float acc[H_];
#pragma unroll
    for (int h = 0; h < H_; ++h) acc[h] = 0.f;
    for (int i = 0; i < IN_; ++i) {
        const float w = nw[(size_t)i * OUT_ + j];
#pragma unroll
        for (int h = 0; h < H_; ++h)
            acc[h] = fmaf(w, W1[h * D_ + OUT_ + i], acc[h]);
    }
    const float bj = bias[j];
    float g1[H_];
#pragma unroll
    for (int h = 0; h < H_; ++h)
        g1[h] = fmaxf(acc[h] + bj * W1[h * D_ + (D_ - 1)] + b1[h], 0.f);
    float s = 0.f;
#pragma unroll
    for (int g = 0; g < H_; ++g) {
        float v = b2[g];
#pragma unroll
        for (int h = 0; h < H_; ++h) v = fmaf(W2[g * H_ + h], g1[h], v);
        s = fmaf(W3[g], fmaxf(v, 0.f), s);
    }
    nb[j] = bj + s + b3[0];
}

// ---------------------------------------------------------------------------
// K4: logits = X @ new_w + new_b  via V_WMMA_F32_16X16X4_F32 (fp32 matrix pipe)
//
// Block tile 128M x 128N, 8 waves = 4 M-waves x 2 N-waves; wave tile 32M x 64N
// (2x4 = 8 accumulators of 8 VGPRs). K staged 32-wide, double-buffered LDS.
// Xs kept [m][k] (A-frag = contiguous K-pair -> 1 ds_load_b64/lane).
// Ws stored TRANSPOSED [n][k] so a B-frag {W[k][n],W[k+1][n]} is contiguous
// -> 1 ds_load_b64/lane, no v_mov pair-assembly. Stride 34 floats: read
// pattern n*34 mod 64 hits 16 distinct banks for 16 lanes (conflict-free).
// ---------------------------------------------------------------------------
#define TM  128
#define TNB 128
#define TK  32
#define NK  (IN_ / TK)
#define XS_STRIDE 36   // float4-store aligned (144B), staggered banks
#define WS_STRIDE 34   // b64-aligned (136B), conflict-free frag reads

__global__ __launch_bounds__(256) void k_gemm(const float* __restrict__ X,
                                              const float* __restrict__ Wn,
                                              const float* __restrict__ nb,
                                              float* __restrict__ out) {
    __shared__ float Xs[2][TM][XS_STRIDE];
    __shared__ float Ws[2][TNB][WS_STRIDE];   // transposed: [n][k]

    const int t    = threadIdx.x;
    const int wave = t >> 5;
    const int lane = t & 31;
    const int lrow = lane & 15;
    const int lhi  = lane >> 4;            // lane half: K-pair {0,1} vs {2,3}
    const int mwv  = (wave & 3) * 32;      // wave M offset in block tile
    const int nwv  = (wave >> 2) * 64;     // wave N offset in block tile
    const int m0   = blockIdx.y * TM;
    const int n0   = blockIdx.x * TNB;

    // cooperative-load thread mapping
    const int mth = t >> 3;                // 0..31  (X tile rows, 4 passes)
    const int kq  = (t & 7) * 4;           // 0..28  (X tile k quad)
    const int kb  = (t >> 5) * 2;          // 0,2,..14 (W tile k pair base)
    const int nq  = (t & 31) * 4;          // 0..124 (W tile n quad)

    v8f c[2][4] = {};
    float4 xg[4];
    float4 wg[2][2];                       // [pp][row k / k+1]

    auto load_tiles = [&](int k0) {
#pragma unroll
        for (int p = 0; p < 4; ++p)
            xg[p] = *(const float4*)(X + (size_t)(m0 + mth + 32 * p) * IN_ + k0 + kq);
#pragma unroll
        for (int pp = 0; pp < 2; ++pp) {
            const int k = kb + pp * 16;
            wg[pp][0] = *(const float4*)(Wn + (size_t)(k0 + k)     * OUT_ + n0 + nq);
            wg[pp][1] = *(const float4*)(Wn + (size_t)(k0 + k + 1) * OUT_ + n0 + nq);
        }
    };
    auto store_tiles = [&](int buf) {
#pragma unroll
        for (int p = 0; p < 4; ++p)
            *(float4*)&Xs[buf][mth + 32 * p][kq] = xg[p];
#pragma unroll
        for (int pp = 0; pp < 2; ++pp) {
            const int k = kb + pp * 16;
            const float r0[4] = { wg[pp][0].x, wg[pp][0].y, wg[pp][0].z, wg[pp][0].w };
            const float r1[4] = { wg[pp][1].x, wg[pp][1].y, wg[pp][1].z, wg[pp][1].w };
#pragma unroll
            for (int cc = 0; cc < 4; ++cc) {
                v2f pair; pair.x = r0[cc]; pair.y = r1[cc];
                *(v2f*)&Ws[buf][nq + cc][k] = pair;     // transposed b64 store
            }
        }
    };

    load_tiles(0);
    store_tiles(0);
    __syncthreads();

    for (int kt = 0; kt < NK; ++kt) {
        const int cur = kt & 1;
        if (kt + 1 < NK) load_tiles((kt + 1) * TK);     // overlap with compute

#pragma unroll
        for (int kk = 0; kk < TK; kk += 4) {
            const int kf = kk + 2 * lhi;
            const v2f a0 = *(const v2f*)&Xs[cur][mwv + lrow][kf];
            const v2f a1 = *(const v2f*)&Xs[cur][mwv + 16 + lrow][kf];
#pragma unroll
            for (int tn = 0; tn < 4; ++tn) {
                const v2f b = *(const v2f*)&Ws[cur][nwv + tn * 16 + lrow][kf];
                c[0][tn] = __builtin_amdgcn_wmma_f32_16x16x4_f32(
                    false, a0, false, b, (short)0, c[0][tn], false, false);
                c[1][tn] = __builtin_amdgcn_wmma_f32_16x16x4_f32(
                    false, a1, false, b, (short)0, c[1][tn], false, false);
            }
        }

        if (kt + 1 < NK) store_tiles(cur ^ 1);
        __syncthreads();
    }

    // epilogue: + new_b, write logits
#pragma unroll
    for (int tn = 0; tn < 4; ++tn) {
        const int n = n0 + nwv + tn * 16 + lrow;
        const float bn = nb[n];
#pragma unroll
        for (int mt = 0; mt < 2; ++mt) {
#pragma unroll
            for (int r = 0; r < 8; ++r) {
                const int m = m0 + mwv + mt * 16 + r + 8 * lhi;
                out[(size_t)m * OUT_ + n] = c[mt][tn][r] + bn;
            }
        }
    }
}

// ---------------------------------------------------------------------------
// K5: in-place row softmax on d_out (2048 rows x 3072)
// ---------------------------------------------------------------------------
__global__ __launch_bounds__(256) void k_softmax(float* __restrict__ out) {
    __shared__ float buf[OUT_];
    __shared__ float red[256];
    const int t = threadIdx.x;
    const size_t base = (size_t)blockIdx.x * OUT_;

    float mx = -3.402823466e38f;
    for (int n = t; n < OUT_; n += 256) {
        const float v = out[base + n];
        buf[n] = v;
        mx = fmaxf(mx, v);
    }
    red[t] = mx;
    __syncthreads();
    for (int s = 128; s > 0; s >>= 1) {
        if (t < s) red[t] = fmaxf(red[t], red[t + s]);
        __syncthreads();
    }
    mx = red[0];
    __syncthreads();

    float sum = 0.f;
    for (int n = t; n < OUT_; n += 256) {
        const float e = expf(buf[n] - mx);
        buf[n] = e;
        sum += e;
    }
    red[t] = sum;
    __syncthreads();
    for (int s = 128; s > 0; s >>= 1) {
        if (t < s) red[t] += red[t + s];
        __syncthreads();
    }
    const float inv = 1.f / red[0];
    for (int n = t; n < OUT_; n += 256) out[base + n] = buf[n] * inv;
}

// ---------------------------------------------------------------------------
extern "C" void kernel_launch(void* const* d_in, const int* in_sizes, int n_in,
                              void* d_out, int out_size, void* d_ws, size_t ws_size,
                              hipStream_t stream) {
    const float* X      = (const float*)d_in[0];
    const float* weight = (const float*)d_in[1];
    const float* bias   = (const float*)d_in[2];
    const float* W1     = (const float*)d_in[3];
    const float* b1     = (const float*)d_in[4];
    const float* W2     = (const float*)d_in[5];
    const float* b2     = (const float*)d_in[6];
    const float* W3     = (const float*)d_in[7];
    const float* b3     = (const float*)d_in[8];

    float* ws = (float*)d_ws;
    float* rp = ws;                         // 3072*10
    float* cb = rp + IN_ * H_;              // 3072*10
    float* nw = cb + OUT_ * H_;             // 3072*3072
    float* nb = nw + (size_t)IN_ * OUT_;    // 3072
    float* out = (float*)d_out;

    k_rowpart<<<dim3(96),        256, 0, stream>>>(weight, W1, rp);
    k_colbias<<<dim3(12),        256, 0, stream>>>(weight, bias, W1, b1, cb);
    k_update <<<dim3(192, 192),  256, 0, stream>>>(weight, rp, cb, W2, b2, W3, b3, nw);
    k_gpath  <<<dim3(12),        256, 0, stream>>>(nw, bias, W1, b1, W2, b2, W3, b3, nb);
    k_gemm   <<<dim3(OUT_ / TNB, B_ / TM), 256, 0, stream>>>(X, nw, nb, out);
    k_softmax<<<dim3(B_),        256, 0, stream>>>(out);
}